// WaveletTransform_16166256902650
// MI455X (gfx1250) — compile-verified
//
#include <hip/hip_runtime.h>
#include <stdint.h>

// Vector types (ext_vector_type so __builtin_nontemporal_store accepts them)
typedef __attribute__((ext_vector_type(4))) float        v4f;
typedef __attribute__((ext_vector_type(4))) unsigned int v4u;
typedef __attribute__((ext_vector_type(8))) int          v8i;
typedef __attribute__((ext_vector_type(4))) int          v4i;

#define IN_W          512
#define IN_H          512
#define ROWS_PER_TILE 8                       // input rows staged per workgroup
#define TILE_ELEMS    (ROWS_PER_TILE * IN_W)  // 4096 floats = 16 KB
#define OUT_W         (IN_W / 2)              // 256
#define OUT_PLANE     (OUT_W * OUT_W)         // 65536

__global__ __launch_bounds__(256) void haar_dwt_kernel(const float* __restrict__ x,
                                                       float* __restrict__ out) {
    __shared__ float tile[TILE_ELEMS];

    const int plane   = blockIdx.y;  // b*C + c, 0..511  (input plane)
    const int tileIdx = blockIdx.x;  // 0..63            (8-row tile within plane)
    const int t       = threadIdx.x;

    // ---- Stage 1: TDM DMA of 16 KB contiguous tile (global -> LDS) ----
    // Only wave 0 issues the tensor op; TENSORcnt is per-wave, so wave 0
    // waits on it, then the workgroup barrier publishes the LDS data.
    if (t < 32) {
        const float* src =
            x + (size_t)plane * (IN_H * IN_W) + (size_t)tileIdx * TILE_ELEMS;
        unsigned long long ga  = (unsigned long long)(uintptr_t)src;
        unsigned int       lds = (unsigned int)(uintptr_t)(&tile[0]); // flat addr low32 == LDS byte offset

        // D# group 0: [1:0]=count=1 | lds_addr | global_addr[56:0] | type=2
        v4u g0;
        g0.x = 1u;
        g0.y = lds;
        g0.z = (unsigned int)(ga & 0xFFFFFFFFull);
        g0.w = (unsigned int)((ga >> 32) & 0x1FFFFFFull) | (2u << 30);

        // D# group 1: data_size=4B, 1-D tile of TILE_ELEMS elements
        v8i g1;
        g1[0] = (2 << 16);                         // data_size = 2 (4 bytes); mask/flags = 0
        g1[1] = (TILE_ELEMS & 0xFFFF) << 16;       // tensor_dim0[15:0]   (bits 63:48)
        g1[2] = (1 << 16) | (TILE_ELEMS >> 16);    // tensor_dim0[31:16] | tensor_dim1=1
        g1[3] = (TILE_ELEMS << 16);                // tensor_dim1 hi=0   | tile_dim0=4096
        g1[4] = 1;                                 // tile_dim1=1, tile_dim2=0
        g1[5] = TILE_ELEMS;                        // tensor_dim0_stride (unused for 1 row)
        g1[6] = 0;
        g1[7] = 0;

        v4i gz4 = {0, 0, 0, 0};                    // groups 2/3 unused (<=2D tensor)
        v8i gz8 = {0, 0, 0, 0, 0, 0, 0, 0};        // extra operand (6-arg form), unused
        __builtin_amdgcn_tensor_load_to_lds(g0, g1, gz4, gz4, gz8, /*cpol=*/0);
        __builtin_amdgcn_s_wait_tensorcnt(0);
    }
    __syncthreads();

    // ---- Stage 2: butterfly transform from LDS, NT stores to 4 planes ----
    const int r  = t >> 6;          // 0..3 : output row within tile
    const int c0 = (t & 63) << 2;   // 0..252 step 4 : first output col (4 blocks/thread)

    const v4f* row0 = (const v4f*)&tile[(2 * r)     * IN_W + (c0 << 1)];
    const v4f* row1 = (const v4f*)&tile[(2 * r + 1) * IN_W + (c0 << 1)];
    v4f a0 = row0[0], a1 = row0[1];   // top row:    p q p q p q p q
    v4f b0 = row1[0], b1 = row1[1];   // bottom row: r s r s r s r s

    v4f ll, lh, hl, hh;
    {
        // block j: p=a[2j], q=a[2j+1], r=b[2j], s=b[2j+1]
        float t0, t1, t2, t3;
        #define HAAR(P, Q, R, S, C)                               \
            t0 = (P) + (Q); t1 = (P) - (Q);                       \
            t2 = (R) + (S); t3 = (R) - (S);                       \
            ll.C = (t0 + t2) * 0.5f; lh.C = (t0 - t2) * 0.5f;     \
            hl.C = (t1 + t3) * 0.5f; hh.C = (t1 - t3) * 0.5f;
        HAAR(a0.x, a0.y, b0.x, b0.y, x)
        HAAR(a0.z, a0.w, b0.z, b0.w, y)
        HAAR(a1.x, a1.y, b1.x, b1.y, z)
        HAAR(a1.z, a1.w, b1.z, b1.w, w)
        #undef HAAR
    }

    const int h2 = tileIdx * (ROWS_PER_TILE / 2) + r;  // global output row 0..255
    float* o = out + (size_t)plane * 4 * OUT_PLANE + (size_t)h2 * OUT_W + c0;

    __builtin_nontemporal_store(ll, (v4f*)(o));                 // channel c*4+0 (LL)
    __builtin_nontemporal_store(lh, (v4f*)(o + OUT_PLANE));     // channel c*4+1 (LH)
    __builtin_nontemporal_store(hl, (v4f*)(o + 2 * OUT_PLANE)); // channel c*4+2 (HL)
    __builtin_nontemporal_store(hh, (v4f*)(o + 3 * OUT_PLANE)); // channel c*4+3 (HH)
}

extern "C" void kernel_launch(void* const* d_in, const int* in_sizes, int n_in,
                              void* d_out, int out_size, void* d_ws, size_t ws_size,
                              hipStream_t stream) {
    const float* x   = (const float*)d_in[0];
    float*       out = (float*)d_out;

    // 512 planes (B*C), 64 eight-row tiles per plane, 256 threads (8 wave32) per WG
    dim3 grid(IN_H / ROWS_PER_TILE, 16 * 32);
    dim3 block(256);
    haar_dwt_kernel<<<grid, block, 0, stream>>>(x, out);
}